// CSML_class_6081673691780
// MI455X (gfx1250) — compile-verified
//
#include <hip/hip_runtime.h>

// MI455X / gfx1250, wave32. One workgroup (128 threads = 4 waves) per batch
// element. Stage all gathered embeddings in LDS, run the bpd,bnd->bpn GEMM
// on V_WMMA_F32_16X16X4_F32 (exact f32), do hinge losses on VALU, reduce
// deterministically (fixed tree, no float atomics).

#define BATCH 1024
#define PP 32
#define NNEG 64
#define NBH 32
#define NNH 32
#define DIM 64
#define STR 68   // LDS row stride in floats: 16B aligned, bank-conflict-free frags

typedef __attribute__((ext_vector_type(2))) float v2f;
typedef __attribute__((ext_vector_type(8))) float v8f;

__global__ __launch_bounds__(128) void csml_triplet_main(
    const float* __restrict__ u2e, const float* __restrict__ v2e,
    const float* __restrict__ margin_uv, const float* __restrict__ margin_vv,
    const float* __restrict__ margin_uu,
    const int* __restrict__ train_u, const int* __restrict__ pos_idx,
    const int* __restrict__ neg_idx, const int* __restrict__ nb_idx,
    const int* __restrict__ nn_idx,
    const unsigned char* __restrict__ pos_mask,
    const unsigned char* __restrict__ neg_mask,
    const unsigned char* __restrict__ nb_mask,
    const unsigned char* __restrict__ nn_mask,
    float* __restrict__ ws)
{
    __shared__ __align__(16) float sP [PP  * STR];
    __shared__ __align__(16) float sN [NNEG* STR];
    __shared__ __align__(16) float sNB[NBH * STR];
    __shared__ __align__(16) float sNN[NNH * STR];
    __shared__ __align__(16) float sPN[PP  * STR];
    __shared__ __align__(16) float sU [DIM];
    __shared__ float sd1[PP], sd2[NNEG], sd1s[NBH], sd2s[NNH];
    __shared__ float sp2[PP], sn2[NNEG], smv[PP];
    __shared__ float spmf[PP], snmf[NNEG], snbmf[NBH], snnmf[NNH];
    __shared__ float sScal[2];
    __shared__ float red[128];

    const int b   = blockIdx.x;
    const int tid = threadIdx.x;

    // ---- gather 161 rows x 64 floats into LDS (float4 segments) ----
    for (int idx = tid; idx < 161 * 16; idx += 128) {
        int row = idx >> 4, seg = idx & 15;
        const float4* src; float* dst;
        if (row < 32)       { src = (const float4*)(v2e + (size_t)pos_idx[b*PP  + row      ] * DIM); dst = sP  + row      * STR; }
        else if (row < 96)  { int r = row - 32;  src = (const float4*)(v2e + (size_t)neg_idx[b*NNEG + r] * DIM); dst = sN  + r * STR; }
        else if (row < 128) { int r = row - 96;  src = (const float4*)(u2e + (size_t)nb_idx [b*NBH  + r] * DIM); dst = sNB + r * STR; }
        else if (row < 160) { int r = row - 128; src = (const float4*)(u2e + (size_t)nn_idx [b*NNH  + r] * DIM); dst = sNN + r * STR; }
        else                { src = (const float4*)(u2e + (size_t)train_u[b] * DIM); dst = sU; }
        ((float4*)dst)[seg] = src[seg];
    }
    if (tid < 32) {
        spmf [tid] = pos_mask[b*PP  + tid] ? 1.f : 0.f;
        snbmf[tid] = nb_mask [b*NBH + tid] ? 1.f : 0.f;
        snnmf[tid] = nn_mask [b*NNH + tid] ? 1.f : 0.f;
        smv  [tid] = margin_vv[pos_idx[b*PP + tid]];
    }
    if (tid < 64) snmf[tid] = neg_mask[b*NNEG + tid] ? 1.f : 0.f;
    if (tid == 0) { int iu = train_u[b]; sScal[0] = margin_uv[iu]; sScal[1] = margin_uu[iu]; }
    __syncthreads();

    // ---- squared distances / norms (256 reduction tasks across 128 threads) ----
    for (int rep = 0; rep < 2; ++rep) {
        int tt = tid + rep * 128;
        if (tt < 32) {
            const float* r = sP + tt * STR; float s = 0.f;
            for (int d = 0; d < DIM; ++d) { float df = sU[d] - r[d]; s += df * df; }
            sd1[tt] = s;
        } else if (tt < 96) {
            int k = tt - 32; const float* r = sN + k * STR; float s = 0.f;
            for (int d = 0; d < DIM; ++d) { float df = sU[d] - r[d]; s += df * df; }
            sd2[k] = s;
        } else if (tt < 128) {
            int i = tt - 96; const float* r = sNB + i * STR; float s = 0.f;
            for (int d = 0; d < DIM; ++d) { float df = sU[d] - r[d]; s += df * df; }
            sd1s[i] = s;
        } else if (tt < 160) {
            int k = tt - 128; const float* r = sNN + k * STR; float s = 0.f;
            for (int d = 0; d < DIM; ++d) { float df = sU[d] - r[d]; s += df * df; }
            sd2s[k] = s;
        } else if (tt < 192) {
            int i = tt - 160; const float* r = sP + i * STR; float s = 0.f;
            for (int d = 0; d < DIM; ++d) s += r[d] * r[d];
            sp2[i] = s;
        } else {
            int k = tt - 192; const float* r = sN + k * STR; float s = 0.f;
            for (int d = 0; d < DIM; ++d) s += r[d] * r[d];
            sn2[k] = s;
        }
    }

    // ---- pn[i][k] = p_e . n_e^T via V_WMMA_F32_16X16X4_F32 ----
    // 8 output tiles of 16x16 (P=32 rows x N=64 cols); 4 waves x 2 tiles each;
    // K=64 -> 16 chained WMMAs per tile. ISA 7.12.2 f32 layouts:
    //   A 16x4: lanes 0-15 hold K={0,1} in v[0],v[1]; lanes 16-31 hold K={2,3}
    //   B  4x16: mirrored (lane = N column)
    //   C/D 16x16: vgpr v, lane L -> row = 8*(L>>4)+v, col = L&15
    {
        const int lane = tid & 31, wave = tid >> 5;
        const int m = lane & 15, h = lane >> 4;
        for (int rep = 0; rep < 2; ++rep) {
            int t  = wave + rep * 4;
            int ti = t >> 2, tk = t & 3;
            const float* arow = sP + (ti * 16 + m) * STR;
            const float* brow = sN + (tk * 16 + m) * STR;
            v8f acc = {};
#pragma unroll
            for (int kk = 0; kk < 16; ++kk) {
                v2f a  = *(const v2f*)(arow + kk * 4 + h * 2);
                v2f bb = *(const v2f*)(brow + kk * 4 + h * 2);
                acc = __builtin_amdgcn_wmma_f32_16x16x4_f32(
                    false, a, false, bb, (short)0, acc, false, false);
            }
            float* orow = sPN + (ti * 16 + h * 8) * STR + tk * 16 + m;
#pragma unroll
            for (int v = 0; v < 8; ++v) orow[v * STR] = acc[v];
        }
    }
    __syncthreads();

    // ---- hinge losses ----
    const float m_u = sScal[0], m_s = sScal[1];
    float acc_uv = 0.f, acc_vv = 0.f, acc_uu = 0.f;
    for (int pair = tid; pair < PP * NNEG; pair += 128) {
        int i = pair >> 6, k = pair & 63;
        float w = spmf[i] * snmf[k];
        acc_uv += w * fmaxf(m_u + sd1[i] - sd2[k], 0.f);
        float d2v  = sp2[i] + sn2[k] - 2.f * sPN[i * STR + k];
        float base = sd1[i] - d2v;
        float inner = 0.f;
#pragma unroll
        for (int j = 0; j < PP; ++j) inner += spmf[j] * fmaxf(smv[j] + base, 0.f);
        acc_vv += w * inner;
    }
    for (int pair = tid; pair < NBH * NNH; pair += 128) {
        int i = pair >> 5, k = pair & 31;
        acc_uu += snbmf[i] * snnmf[k] * fmaxf(m_s + sd1s[i] - sd2s[k], 0.f);
    }
    red[tid] = acc_uv + 1.0f * acc_vv + 1.0f * acc_uu;  // W1=W3=1
    __syncthreads();
    for (int s = 64; s > 0; s >>= 1) {
        if (tid < s) red[tid] += red[tid + s];
        __syncthreads();
    }
    if (tid == 0) {
        ws[0 * BATCH + b] = red[0];
        ws[1 * BATCH + b] = m_u;
        float s2 = 0.f, s3 = 0.f;
        for (int j = 0; j < PP; ++j) { s2 += smv[j] * spmf[j]; s3 += spmf[j]; }
        ws[2 * BATCH + b] = s2;
        ws[3 * BATCH + b] = s3;
        ws[4 * BATCH + b] = m_s;
    }
}

__global__ __launch_bounds__(256) void csml_triplet_finalize(
    const float* __restrict__ ws, float* __restrict__ out)
{
    __shared__ float sh[5][256];
    const int tid = threadIdx.x;
    for (int a = 0; a < 5; ++a) {
        float s = 0.f;
        for (int j = 0; j < 4; ++j) s += ws[a * BATCH + tid * 4 + j];
        sh[a][tid] = s;
    }
    __syncthreads();
    for (int st = 128; st > 0; st >>= 1) {
        if (tid < st)
            for (int a = 0; a < 5; ++a) sh[a][tid] += sh[a][tid + st];
        __syncthreads();
    }
    if (tid == 0) {
        float hinge   = sh[0][0];
        float loss_am = sh[1][0] / (float)BATCH
                      + sh[2][0] / sh[3][0]
                      + sh[4][0] / (float)BATCH;
        out[0] = hinge + 0.1f * loss_am;   // W2 = 0.1
    }
}

extern "C" void kernel_launch(void* const* d_in, const int* in_sizes, int n_in,
                              void* d_out, int out_size, void* d_ws, size_t ws_size,
                              hipStream_t stream) {
    (void)in_sizes; (void)n_in; (void)out_size; (void)ws_size;
    const float* u2e       = (const float*)d_in[0];
    const float* v2e       = (const float*)d_in[1];
    const float* margin_uv = (const float*)d_in[2];
    const float* margin_vv = (const float*)d_in[3];
    const float* margin_uu = (const float*)d_in[4];
    const int*   train_u   = (const int*)d_in[5];
    const int*   pos_idx   = (const int*)d_in[6];
    const int*   neg_idx   = (const int*)d_in[7];
    const int*   nb_idx    = (const int*)d_in[8];
    const int*   nn_idx    = (const int*)d_in[9];
    const unsigned char* pos_mask = (const unsigned char*)d_in[10];
    const unsigned char* neg_mask = (const unsigned char*)d_in[11];
    const unsigned char* nb_mask  = (const unsigned char*)d_in[12];
    const unsigned char* nn_mask  = (const unsigned char*)d_in[13];
    float* ws  = (float*)d_ws;
    float* out = (float*)d_out;

    csml_triplet_main<<<BATCH, 128, 0, stream>>>(
        u2e, v2e, margin_uv, margin_vv, margin_uu,
        train_u, pos_idx, neg_idx, nb_idx, nn_idx,
        pos_mask, neg_mask, nb_mask, nn_mask, ws);
    csml_triplet_finalize<<<1, 256, 0, stream>>>(ws, out);
}